// BBoxGuidedConceptLoss_48765058679411
// MI455X (gfx1250) — compile-verified
//
#include <hip/hip_runtime.h>
#include <hip/hip_bf16.h>

// Problem constants (match reference)
#define BB     64
#define KK     128
#define HH     64
#define WW     64
#define HW     4096           // H*W
#define NROWS  8192           // B*K
#define NBOX   256
#define ALPHA_ 1.0f
#define BETA_  0.5f

typedef __attribute__((ext_vector_type(2))) float v2f;
typedef __attribute__((ext_vector_type(8))) float v8f;

__device__ __forceinline__ float waveMax(float v) {
#pragma unroll
    for (int o = 16; o > 0; o >>= 1) v = fmaxf(v, __shfl_xor(v, o, 32));
    return v;
}

__device__ __forceinline__ float waveSum(float v) {
#pragma unroll
    for (int o = 16; o > 0; o >>= 1) v += __shfl_xor(v, o, 32);
    return v;
}

// Fast sigmoid: v_exp_f32 + v_rcp_f32 (~1 ulp), no IEEE division chain.
__device__ __forceinline__ float fast_sigmoid(float v) {
    const float e = __expf(-v);                    // v_mul + v_exp_f32
    return __builtin_amdgcn_rcpf(1.0f + e);        // v_add + v_rcp_f32
}

// ---------------------------------------------------------------------------
// Kernel 1: row-max over 4096 floats per (b,k) row.  Pure HBM streaming:
// 128 MB total -> ~5.5 us at 23.3 TB/s.  256 threads/block (8 waves),
// each thread does 4x float4 coalesced loads.
// ---------------------------------------------------------------------------
__global__ void k_rowmax(const float* __restrict__ cams, float* __restrict__ logits) {
    const int row = blockIdx.x;
    const int t   = threadIdx.x;
    const float4* p = (const float4*)(cams + (size_t)row * HW);

    float m = -3.402823466e+38f;
#pragma unroll
    for (int c = 0; c < 4; ++c) {
        float4 v = p[c * 256 + t];
        m = fmaxf(m, fmaxf(fmaxf(v.x, v.y), fmaxf(v.z, v.w)));
    }
    m = waveMax(m);

    __shared__ float red[8];
    const int wave = t >> 5, lane = t & 31;
    if (lane == 0) red[wave] = m;
    __syncthreads();
    if (t == 0) {
        float mm = red[0];
#pragma unroll
        for (int i = 1; i < 8; ++i) mm = fmaxf(mm, red[i]);
        logits[row] = mm;
    }
}

// ---------------------------------------------------------------------------
// Kernel 2: numerically stable BCE-with-logits sum over 8192 elements.
// Single block, deterministic tree reduction.  (Keeps full-precision expf/
// log1pf here: only 8192 elements, and cls_loss accuracy matters most.)
// ---------------------------------------------------------------------------
__global__ void k_bce(const float* __restrict__ logits,
                      const float* __restrict__ gt,
                      float* __restrict__ cls_sum) {
    const int t = threadIdx.x;
    float s = 0.0f;
#pragma unroll
    for (int j = 0; j < NROWS / 256; ++j) {
        const int i = t + j * 256;
        const float z = logits[i];
        const float y = gt[i];
        // max(z,0) - z*y + log1p(exp(-|z|))
        s += fmaxf(z, 0.0f) - z * y + log1pf(expf(-fabsf(z)));
    }
    s = waveSum(s);
    __shared__ float red[8];
    const int wave = t >> 5, lane = t & 31;
    if (lane == 0) red[wave] = s;
    __syncthreads();
    if (t == 0) {
        float tot = 0.0f;
#pragma unroll
        for (int i = 0; i < 8; ++i) tot += red[i];
        cls_sum[0] = tot;
    }
}

// ---------------------------------------------------------------------------
// Kernel 3: per-box localization loss.  One block (8 waves) per box.
// Each wave owns a contiguous 512-pixel segment; per iteration each lane
// contributes 2 pixels to a 16x4 f32 A-matrix and V_WMMA_F32_16X16X4_F32
// with an all-ones B accumulates the full-precision sum into the f32
// accumulator tile: D[m][n] = sum_k A[m][k] + C[m][n].  Two accumulator
// chains: inside = (s-1)^2*mask, outside = s^2*(1-mask).
// EXEC is all-ones throughout (no divergence before/around the WMMAs).
// ---------------------------------------------------------------------------
__global__ void k_box(const float* __restrict__ cams,
                      const int* __restrict__ box_b, const int* __restrict__ box_c,
                      const int* __restrict__ y0, const int* __restrict__ y1,
                      const int* __restrict__ x0, const int* __restrict__ x1,
                      float* __restrict__ loc_out) {
    const int n = blockIdx.x;
    const int t = threadIdx.x, wave = t >> 5, lane = t & 31;

    const int b  = box_b[n], c = box_c[n];
    const int ry0 = y0[n], ry1 = y1[n], rx0 = x0[n], rx1 = x1[n];
    const float* base = cams + ((size_t)b * KK + c) * HW;

    v8f accIn  = {};
    v8f accOut = {};
    const v2f bones = {1.0f, 1.0f};

    const int segBase = wave * 512;
#pragma unroll
    for (int it = 0; it < 8; ++it) {
        const int idx = segBase + it * 64 + lane * 2;
        v2f val = *(const v2f*)(base + idx);

        v2f ain, aout;
#pragma unroll
        for (int e = 0; e < 2; ++e) {
            const int p  = idx + e;
            const int py = p >> 6;       // W = 64
            const int px = p & 63;
            const float sg = fast_sigmoid(val[e]);
            const bool inBox = (py >= ry0) & (py < ry1) & (px >= rx0) & (px < rx1);
            const float d = sg - 1.0f;
            ain[e]  = inBox ? d * d : 0.0f;
            aout[e] = inBox ? 0.0f : sg * sg;
        }
        // D = A * ones + C  (full f32 precision sum-reduction)
        accIn  = __builtin_amdgcn_wmma_f32_16x16x4_f32(false, ain,  false, bones,
                                                       (short)0, accIn,  false, false);
        accOut = __builtin_amdgcn_wmma_f32_16x16x4_f32(false, aout, false, bones,
                                                       (short)0, accOut, false, false);
    }

    // Lane-half 0 holds rows 0-7, half 1 rows 8-15 (columns identical since
    // B is all-ones): sum the 8 accumulator VGPRs, then add across halves.
    float pin  = 0.0f, pout = 0.0f;
#pragma unroll
    for (int i = 0; i < 8; ++i) { pin += accIn[i]; pout += accOut[i]; }
    pin  += __shfl_xor(pin, 16, 32);
    pout += __shfl_xor(pout, 16, 32);

    __shared__ float sIn[8], sOut[8];
    if (lane == 0) { sIn[wave] = pin; sOut[wave] = pout; }
    __syncthreads();
    if (t == 0) {
        float tin = 0.0f, tout = 0.0f;
#pragma unroll
        for (int i = 0; i < 8; ++i) { tin += sIn[i]; tout += sOut[i]; }
        const float area    = (float)((ry1 - ry0) * (rx1 - rx0));
        const float inside  = tin  / (area + 1e-6f);
        const float outside = tout / ((float)HW - area + 1e-6f);
        loc_out[n] = inside + outside;
    }
}

// ---------------------------------------------------------------------------
// Kernel 4: final scalar combine (deterministic tree reduce of 256 box losses).
// ---------------------------------------------------------------------------
__global__ void k_final(const float* __restrict__ cls_sum,
                        const float* __restrict__ loc,
                        float* __restrict__ out) {
    const int t = threadIdx.x;
    float v = loc[t];
    v = waveSum(v);
    __shared__ float red[8];
    const int wave = t >> 5, lane = t & 31;
    if (lane == 0) red[wave] = v;
    __syncthreads();
    if (t == 0) {
        float tot = 0.0f;
#pragma unroll
        for (int i = 0; i < 8; ++i) tot += red[i];
        out[0] = ALPHA_ * (cls_sum[0] / (float)NROWS) + BETA_ * (tot / (float)NBOX);
    }
}

// ---------------------------------------------------------------------------
extern "C" void kernel_launch(void* const* d_in, const int* in_sizes, int n_in,
                              void* d_out, int out_size, void* d_ws, size_t ws_size,
                              hipStream_t stream) {
    const float* cams = (const float*)d_in[0];
    const float* gt   = (const float*)d_in[1];
    const int* box_b  = (const int*)d_in[2];
    const int* box_c  = (const int*)d_in[3];
    const int* y0     = (const int*)d_in[4];
    const int* y1     = (const int*)d_in[5];
    const int* x0     = (const int*)d_in[6];
    const int* x1     = (const int*)d_in[7];

    float* ws      = (float*)d_ws;
    float* logits  = ws;               // 8192 floats
    float* cls_sum = ws + NROWS;       // 1 float
    float* loc     = ws + NROWS + 64;  // 256 floats (padded offset)
    float* out     = (float*)d_out;

    k_rowmax<<<NROWS, 256, 0, stream>>>(cams, logits);
    k_bce   <<<1,     256, 0, stream>>>(logits, gt, cls_sum);
    k_box   <<<NBOX,  256, 0, stream>>>(cams, box_b, box_c, y0, y1, x0, x1, loc);
    k_final <<<1,     256, 0, stream>>>(cls_sum, loc, out);
}